// K_attention_MH_12086037971049
// MI455X (gfx1250) — compile-verified
//
#include <hip/hip_runtime.h>

// K_attention_MH: per-token head-pairwise Gaussian kernel mix.
//   xh (16x64 f32) per token; G = xh*xh^T; dist2 = diag_m + diag_n - 2G;
//   K = exp(-dist2*sigma); y = xh + K @ xh.
// Memory-bound (AI ~8 FLOP/B, 33.5MB total traffic, ~1.4us @ 23.3TB/s):
// single fused pass, f32 WMMA (V_WMMA_F32_16X16X4_F32) for both matmuls.

typedef float v2f __attribute__((ext_vector_type(2)));
typedef float v8f __attribute__((ext_vector_type(8)));

#define NHEAD 16
#define HDIM  64
#define TOKC  (NHEAD * HDIM)   // 1024 floats per token
#define WPB   8                // waves per block (256 threads)

__global__ __launch_bounds__(256) void
K_attention_MH_kernel(const float* __restrict__ x,
                      const float* __restrict__ r_sigma,
                      float* __restrict__ y,
                      int num_tokens) {
    // Per-wave private LDS slices: 16x64 x tile + 16x16 K matrix.
    __shared__ float lds_x[WPB][TOKC];       // 8 * 4KB = 32KB
    __shared__ float lds_k[WPB][NHEAD * NHEAD]; // 8 * 1KB = 8KB

    const int lane  = threadIdx.x & 31;
    const int wave  = threadIdx.x >> 5;
    const int token = blockIdx.x * WPB + wave;
    if (token >= num_tokens) return;          // wave-uniform: EXEC stays all-1s

    const float sigma = r_sigma[0];
    const float* xt = x + (size_t)token * TOKC;
    float*       yt = y + (size_t)token * TOKC;
    float* lx = lds_x[wave];
    float* lk = lds_k[wave];

    const int row = lane & 15;   // M (or N) index for this lane
    const int hi  = lane >> 4;   // 0: lanes 0-15, 1: lanes 16-31

    // ---- Phase 1: global -> LDS, 8 x float4 per lane, fully coalesced ----
    #pragma unroll
    for (int i = 0; i < 8; ++i) {
        const int idx = (i * 32 + lane) * 4;           // float index
        const float4 v = *(const float4*)(xt + idx);
        *(float4*)(lx + idx) = v;
    }
    asm volatile("" ::: "memory");

    // ---- Phase 2: Gram matrix G = xh * xh^T  (16 WMMA f32 16x16x4) ----
    // A 16x4 layout: lane m<16 holds K=0,1; lane m+16 holds K=2,3.
    // B 4x16 layout of xh^T chunk is the SAME register image -> reuse a.
    v8f acc = {};
    #pragma unroll
    for (int c = 0; c < 16; ++c) {
        const int off = row * HDIM + c * 4 + hi * 2;
        v2f a;
        a.x = lx[off + 0];
        a.y = lx[off + 1];
        acc = __builtin_amdgcn_wmma_f32_16x16x4_f32(
            false, a, false, a, (short)0, acc, false, false);
    }

    // ---- Phase 3: spill G to LDS (C-layout: vgpr r -> m = r + hi*8, n = row)
    #pragma unroll
    for (int r = 0; r < 8; ++r)
        lk[(r + hi * 8) * NHEAD + row] = acc[r];
    asm volatile("" ::: "memory");

    // ---- Phase 4: read diagonal (the squared norms), form K = exp(-d*sigma)
    float dn = lk[row * NHEAD + row];                  // ||xh[n]||^2
    float dm[8];
    #pragma unroll
    for (int r = 0; r < 8; ++r) {
        const int m = r + hi * 8;
        dm[r] = lk[m * NHEAD + m];                     // ||xh[m]||^2
    }
    #pragma unroll
    for (int r = 0; r < 8; ++r) {
        const float d = dm[r] + dn - 2.0f * acc[r];    // dist^2 >= 0
        lk[(r + hi * 8) * NHEAD + row] = __expf(-d * sigma);
    }
    asm volatile("" ::: "memory");

    // ---- Phase 5: y = xh + K @ xh  (4 col-blocks x 4 K-chunks WMMA) ----
    // Preload K in A-layout: lane m=row holds cols 4c+hi*2, 4c+hi*2+1.
    v2f ka[4];
    #pragma unroll
    for (int c = 0; c < 4; ++c) {
        const int off = row * NHEAD + c * 4 + hi * 2;
        ka[c].x = lk[off + 0];
        ka[c].y = lk[off + 1];
    }

    #pragma unroll
    for (int nb = 0; nb < 4; ++nb) {
        // Residual init: C = xh in C-layout.
        v8f cy;
        #pragma unroll
        for (int r = 0; r < 8; ++r)
            cy[r] = lx[(r + hi * 8) * HDIM + nb * 16 + row];

        #pragma unroll
        for (int c = 0; c < 4; ++c) {
            // B 4x16 chunk: rows g = 4c..4c+3 of xh, cols nb*16..nb*16+15.
            // vgpr0: row 4c+hi*2, vgpr1: row 4c+hi*2+1, N = row.
            const int g0 = c * 4 + hi * 2;
            v2f b;
            b.x = lx[(g0 + 0) * HDIM + nb * 16 + row];
            b.y = lx[(g0 + 1) * HDIM + nb * 16 + row];
            cy = __builtin_amdgcn_wmma_f32_16x16x4_f32(
                false, ka[c], false, b, (short)0, cy, false, false);
        }

        // Store: element (m, nb*16+row), lanes 0-15 write 64B runs.
        #pragma unroll
        for (int r = 0; r < 8; ++r)
            yt[(r + hi * 8) * HDIM + nb * 16 + row] = cy[r];
    }
}

extern "C" void kernel_launch(void* const* d_in, const int* in_sizes, int n_in,
                              void* d_out, int out_size, void* d_ws, size_t ws_size,
                              hipStream_t stream) {
    const float* x       = (const float*)d_in[0];
    const float* r_sigma = (const float*)d_in[1];
    float*       y       = (float*)d_out;

    const int num_tokens = in_sizes[0] / TOKC;   // B*T = 4096
    const int blocks     = (num_tokens + WPB - 1) / WPB;

    hipLaunchKernelGGL(K_attention_MH_kernel,
                       dim3(blocks), dim3(256), 0, stream,
                       x, r_sigma, y, num_tokens);
}